// critical_points_44598940401963
// MI455X (gfx1250) — compile-verified
//
#include <hip/hip_runtime.h>
#include <math.h>

// Problem constants (reference: B=32, C=1024, N=4096, K=3)
#define BB 32
#define CC 1024
#define NN 4096

typedef __attribute__((ext_vector_type(2))) float v2f;
typedef __attribute__((ext_vector_type(8))) float v8f;

// ---------------------------------------------------------------------------
// Kernel 1: per-batch  W(1024x3) @ x[b](3x4096) -> row argmax -> histogram.
// One workgroup per batch, 256 threads = 8 wave32s.
// GEMM via V_WMMA_F32_16X16X4_F32 (K=4, zero-padded K slot), f32 end to end
// so argmax winners match the f32 reference exactly.
// LDS (dynamic, 80 KB): x staged point-major [N][4] with K=3 zero-padded so
// each lane's B fragment is ONE unconditional conflict-free ds_load_b64
// (lower half-wave reads dwords 4n+0..1, upper half 4n+2..3 -> disjoint banks)
// ---------------------------------------------------------------------------
__global__ __launch_bounds__(256) void gemm_argmax_hist_kernel(
    const float* __restrict__ x,      // [B,3,N]
    const float* __restrict__ W,      // [C,3]
    float* __restrict__ counts)       // [B,N] out
{
  extern __shared__ char smem_raw[];
  float* xs4 = (float*)smem_raw;                       // [NN][4]  64 KB
  int*   hist = (int*)(smem_raw + 4u * NN * sizeof(float)); // [NN] 16 KB

  const int b = blockIdx.x;
  const int t = threadIdx.x;

  // Stage x[b] (row-major [3][N]) into padded point-major [N][4]; pad = 0.
  {
    const float* xb = x + (size_t)b * 3 * NN;
    for (int i = t; i < 3 * NN; i += 256) {
      const int d = i >> 12;          // / NN
      const int n = i & (NN - 1);     // % NN
      xs4[(n << 2) | d] = xb[i];
    }
    for (int n = t; n < NN; n += 256) xs4[(n << 2) | 3] = 0.0f;
    for (int i = t; i < NN; i += 256) hist[i] = 0;
  }
  __syncthreads();

  const int lane = t & 31;
  const int wave = t >> 5;
  const int hl   = lane >> 4;   // K-half: 0 -> K=0,1 ; 1 -> K=2,(3=pad)
  const int nl   = lane & 15;   // M (for A) / N (for B,D) position

  // Fixed per-lane B base: fragment for point-tile nt lives at bp + nt*64.
  const float* bp = xs4 + (nl << 2) + (hl << 1);

  // 64 channel tiles of 16; 8 waves -> 8 tiles per wave.
  for (int ct = wave; ct < CC / 16; ct += 8) {
    // A-matrix: 16x4 tile of W, K=3 padded with 0.
    const int ch = ct * 16 + nl;
    v2f a;
    a.x = hl ? W[ch * 3 + 2] : W[ch * 3 + 0];
    a.y = hl ? 0.0f          : W[ch * 3 + 1];

    float vmax[8];
    int   vidx[8];
#pragma unroll
    for (int v = 0; v < 8; ++v) { vmax[v] = -1e30f; vidx[v] = 0; }

    // Sweep all 256 point tiles of 16; keep running per-lane (max, argmax).
#pragma unroll 4
    for (int nt = 0; nt < NN / 16; ++nt) {
      const v2f bm = *(const v2f*)(bp + nt * 64);   // one ds_load_b64
      v8f c = {};
      v8f d = __builtin_amdgcn_wmma_f32_16x16x4_f32(
          /*neg_a=*/false, a, /*neg_b=*/false, bm,
          /*c_mod=*/(short)0, c, /*reuse_a=*/false, /*reuse_b=*/false);
      const int n = nt * 16 + nl;
#pragma unroll
      for (int v = 0; v < 8; ++v) {
        // strict > : first max wins (matches jnp.argmax)
        if (d[v] > vmax[v]) { vmax[v] = d[v]; vidx[v] = n; }
      }
    }

    // Branchless (max, argmax) butterfly across each 16-lane half; ties ->
    // lowest n. xor masks 8..1 stay inside each half.
#pragma unroll
    for (int v = 0; v < 8; ++v) {
      float mv = vmax[v];
      int   mi = vidx[v];
#pragma unroll
      for (int off = 8; off >= 1; off >>= 1) {
        const float ov = __shfl_xor(mv, off, 32);
        const int   oi = __shfl_xor(mi, off, 32);
        const int take = (int)(ov > mv) | ((int)(ov == mv) & (int)(oi < mi));
        mv = take ? ov : mv;
        mi = take ? oi : mi;
      }
      // channel m = v + 8*hl of this tile; one leader lane per half records it
      if (nl == 0) atomicAdd(&hist[mi], 1);
    }
  }
  __syncthreads();

  for (int i = t; i < NN; i += 256)
    counts[(size_t)b * NN + i] = (float)hist[i];
}

// ---------------------------------------------------------------------------
// Kernel 2: per-batch stable ascending argsort of counts (composite-key LDS
// bitonic sort: key = count<<12 | n, all keys distinct => stability for free)
// plus the normalized-counts entropy. 1024 threads per batch.
// ---------------------------------------------------------------------------
__global__ __launch_bounds__(1024) void sort_entropy_kernel(
    const float* __restrict__ counts,  // [B,N]
    float* __restrict__ imp2,          // [B,N] out (indices as float)
    float* __restrict__ ent)           // [B]   out
{
  __shared__ unsigned key[NN];   // 16 KB
  __shared__ float    red[1024]; // 4 KB

  const int b = blockIdx.x;
  const int t = threadIdx.x;
  const float* cb = counts + (size_t)b * NN;

  // sum(counts + eps) = C + N*eps exactly (each channel contributes one count)
  const float invS = 1.0f / (1024.0f + (float)NN * 1e-6f);
  float s = 0.0f;
#pragma unroll
  for (int i = 0; i < 4; ++i) {
    const int n = t + i * 1024;
    const float cf = cb[n];
    const unsigned cu = (unsigned)(cf + 0.5f);     // exact small int
    key[n] = (cu << 12) | (unsigned)n;
    const float p = (cf + 1e-6f) * invS;
    s += p * __log2f(p);
  }
  __syncthreads();

  // Bitonic sort, ascending, 4096 elements / 1024 threads (4 per step).
  for (int k = 2; k <= NN; k <<= 1) {
    for (int j = k >> 1; j > 0; j >>= 1) {
#pragma unroll
      for (int e = 0; e < 4; ++e) {
        const int i = t + e * 1024;
        const int ixj = i ^ j;
        if (ixj > i) {
          const unsigned a = key[i];
          const unsigned c = key[ixj];
          const bool up = ((i & k) == 0);
          if ((a > c) == up) { key[i] = c; key[ixj] = a; }
        }
      }
      __syncthreads();
    }
  }

#pragma unroll
  for (int i = 0; i < 4; ++i) {
    const int n = t + i * 1024;
    imp2[(size_t)b * NN + n] = (float)(key[n] & 0xFFFu);
  }

  // Entropy tree-reduce; ent = -sum(p*log2 p) / log2(4096)
  red[t] = s;
  __syncthreads();
  for (int off = 512; off > 0; off >>= 1) {
    if (t < off) red[t] += red[t + off];
    __syncthreads();
  }
  if (t == 0) ent[b] = -red[0] * (1.0f / 12.0f);
}

// ---------------------------------------------------------------------------
// Outputs, concatenated flat: x[32*3*4096], counts[32*4096], imp2[32*4096],
// ent[32].
// ---------------------------------------------------------------------------
extern "C" void kernel_launch(void* const* d_in, const int* in_sizes, int n_in,
                              void* d_out, int out_size, void* d_ws, size_t ws_size,
                              hipStream_t stream) {
  (void)in_sizes; (void)n_in; (void)out_size; (void)d_ws; (void)ws_size;
  const float* x = (const float*)d_in[0];   // [32,3,4096]
  const float* W = (const float*)d_in[1];   // [1024,3]

  float* out_x      = (float*)d_out;
  float* out_counts = out_x + (size_t)BB * 3 * NN;   // +393216
  float* out_imp2   = out_counts + (size_t)BB * NN;  // +131072
  float* out_ent    = out_imp2 + (size_t)BB * NN;    // +131072

  // Output 0 is the input passthrough.
  hipMemcpyAsync(out_x, x, (size_t)BB * 3 * NN * sizeof(float),
                 hipMemcpyDeviceToDevice, stream);

  const size_t lds_bytes = (size_t)(4 * NN) * sizeof(float)  // padded x: 64 KB
                         + (size_t)NN * sizeof(int);         // hist:     16 KB
  gemm_argmax_hist_kernel<<<BB, 256, lds_bytes, stream>>>(x, W, out_counts);
  sort_entropy_kernel<<<BB, 1024, 0, stream>>>(out_counts, out_imp2, out_ent);
}